// TaskAlignedAssigner_12773232738579
// MI455X (gfx1250) — compile-verified
//
#include <hip/hip_runtime.h>
#include <cstdint>
#include <cstddef>

// ---------------------------------------------------------------------------
// TaskAlignedAssigner for MI455X (gfx1250, wave32).
// Strategy: recompute CIoU per phase (VALU is cheap, HBM/L2 traffic is not),
// use V_WMMA_F32_16X16X4_F32 for the exact one-hot score "gather-as-matmul",
// and async global->LDS (CDNA5 ASYNCcnt path) to stage GT tables.
// ---------------------------------------------------------------------------

#define NMB   64
#define NCLS  80
#define TOPK  13
#define INV_PI2_X4 0.405284734569351f   /* 4 / pi^2 */

typedef __attribute__((ext_vector_type(2))) float v2f;
typedef __attribute__((ext_vector_type(8))) float v8f;

#if defined(__AMDGCN__) && __has_builtin(__builtin_amdgcn_wmma_f32_16x16x4_f32)
#define HAVE_WMMA_F32 1
#else
#define HAVE_WMMA_F32 0
#endif

#define USE_ASYNC_LDS 1

// Reference _ciou (box1 = gt, box2 = pd), eps = 1e-7
__device__ __forceinline__ float ciou_box(float gx1, float gy1, float gx2, float gy2,
                                          float px1, float py1, float px2, float py2) {
    const float eps = 1e-7f;
    float w1 = gx2 - gx1, h1 = gy2 - gy1 + eps;
    float w2 = px2 - px1, h2 = py2 - py1 + eps;
    float iw = fmaxf(fminf(gx2, px2) - fmaxf(gx1, px1), 0.f);
    float ih = fmaxf(fminf(gy2, py2) - fmaxf(gy1, py1), 0.f);
    float inter = iw * ih;
    float uni = w1 * h1 + w2 * h2 - inter + eps;
    float iou = inter / uni;
    float cw = fmaxf(gx2, px2) - fminf(gx1, px1);
    float ch = fmaxf(gy2, py2) - fminf(gy1, py1);
    float c2 = cw * cw + ch * ch + eps;
    float dx = px1 + px2 - gx1 - gx2;
    float dy = py1 + py2 - gy1 - gy2;
    float rho2 = 0.25f * (dx * dx + dy * dy);
    float da = atanf(w2 / h2) - atanf(w1 / h1);
    float v = INV_PI2_X4 * da * da;
    float alpha = v / (v - iou + (1.f + eps));
    return iou - (rho2 / c2 + v * alpha);
}

// ---------------------------------------------------------------------------
// Kernel A: bbox_scores[b,g,a] = pd_scores[b,a,label[b,g]] via FP32 WMMA.
// D(16x16) = onehot(16x80) x pd_scores_chunk(80x16), K accumulated 4 at a time.
// Exact: A is one-hot in fp32, so the accumulation adds exactly one product.
// One wave -> one (b, gTile, aTile) 16x16 tile.
// ---------------------------------------------------------------------------
__global__ void scores_wmma_kernel(const float* __restrict__ pdScores,
                                   const int* __restrict__ gtLabels,
                                   float* __restrict__ scoresMat,
                                   int na, int bs) {
    const int wavesPerBlock = blockDim.x >> 5;
    const int wave = blockIdx.x * wavesPerBlock + (threadIdx.x >> 5);
    const int aTiles = na >> 4;                 // na / 16
    const int gTiles = NMB >> 4;                // 4
    const int tilesPerB = gTiles * aTiles;
    const int b = wave / tilesPerB;
    if (b >= bs) return;
    const int rem = wave % tilesPerB;
    const int gt = rem / aTiles;
    const int at = rem % aTiles;

    const int lane = threadIdx.x & 31;
    const int n  = lane & 15;   // anchor within tile / M row within tile
    const int hi = lane >> 4;   // half-wave selector

    const int myLbl = gtLabels[b * NMB + gt * 16 + n];  // label for A-row M=n
    const int aIdx = at * 16 + n;
    const float* srow = pdScores + ((size_t)b * na + aIdx) * NCLS;

    v8f acc = {};
#if HAVE_WMMA_F32
    for (int kb = 0; kb < NCLS; kb += 4) {
        // A 16x4 layout: lanes0-15 M=lane {K0,K1}; lanes16-31 M=lane-16 {K2,K3}
        const int k0 = kb + hi * 2;
        v2f afrag, bfrag;
        afrag.x = (myLbl == k0)     ? 1.f : 0.f;
        afrag.y = (myLbl == k0 + 1) ? 1.f : 0.f;
        // B 4x16: lane holds column N=n, rows {k0, k0+1}
        bfrag.x = srow[k0];
        bfrag.y = srow[k0 + 1];
        acc = __builtin_amdgcn_wmma_f32_16x16x4_f32(false, afrag, false, bfrag,
                                                    (short)0, acc, false, false);
    }
#else
    // Exact scalar equivalent (gather): D[r] = pd_scores[a=n, label[g=r+hi*8]]
    #pragma unroll
    for (int rr = 0; rr < 8; ++rr) {
        int g = gt * 16 + rr + hi * 8;
        acc[rr] = srow[gtLabels[b * NMB + g]];
    }
#endif
    // D layout: VGPR r -> M = r + hi*8, N = n
    #pragma unroll
    for (int rr = 0; rr < 8; ++rr) {
        int g = gt * 16 + rr + hi * 8;
        scoresMat[((size_t)b * NMB + g) * na + aIdx] = acc[rr];
    }
}

__device__ __forceinline__ unsigned long long umax64(unsigned long long a,
                                                     unsigned long long b) {
    return a > b ? a : b;
}

// ---------------------------------------------------------------------------
// Kernel B: per (b,g) top-13 of align_metric over anchors; scatter mask_pos.
// Key = (metric_bits << 32) | (~anchor): max-order == (value desc, idx asc),
// matching lax.top_k tie-breaking (metric >= 0 so float bits are monotone).
// ---------------------------------------------------------------------------
__global__ void topk_kernel(const float* __restrict__ pdScores,
                            const float* __restrict__ pdBoxes,
                            const float* __restrict__ anc,
                            const int* __restrict__ gtLabels,
                            const float* __restrict__ gtBoxes,
                            const float* __restrict__ maskGt,
                            const float* __restrict__ scoresMat, // may be null
                            unsigned char* __restrict__ maskPos,
                            int na) {
    const int bg = blockIdx.x;            // b*NMB + g
    const int b = bg / NMB;
    const int tid = threadIdx.x;

    __shared__ float sBox[4];
    __shared__ int   sLbl;
    __shared__ float sMg;
    __shared__ unsigned long long sKeys[256];
    __shared__ int sSel[TOPK];

    if (tid == 0) {
        sBox[0] = gtBoxes[(size_t)bg * 4 + 0];
        sBox[1] = gtBoxes[(size_t)bg * 4 + 1];
        sBox[2] = gtBoxes[(size_t)bg * 4 + 2];
        sBox[3] = gtBoxes[(size_t)bg * 4 + 3];
        sLbl = gtLabels[bg];
        sMg  = maskGt[bg];
    }
    __syncthreads();
    if (sMg <= 0.f) return;   // row fully masked -> mask_pos stays 0 (matches ref)

    const float gx1 = sBox[0], gy1 = sBox[1], gx2 = sBox[2], gy2 = sBox[3];
    const int lbl = sLbl;

    unsigned long long keys[TOPK];
    #pragma unroll
    for (int i = 0; i < TOPK; ++i) keys[i] = 0ull;

    for (int a = tid; a < na; a += 256) {
        float x = anc[2 * a], y = anc[2 * a + 1];
        float d = fminf(fminf(x - gx1, y - gy1), fminf(gx2 - x, gy2 - y));
        float metric = 0.f;
        if (d > 1e-9f) {
            const float* pb = pdBoxes + ((size_t)b * na + a) * 4;
            float ci = ciou_box(gx1, gy1, gx2, gy2, pb[0], pb[1], pb[2], pb[3]);
            float ov = fmaxf(ci, 0.f);
            float sc = scoresMat ? scoresMat[(size_t)bg * na + a]
                                 : pdScores[((size_t)b * na + a) * NCLS + lbl];
            float ov2 = ov * ov;
            metric = sc * (ov2 * ov2 * ov2);   // ALPHA=1, BETA=6
        }
        unsigned long long key =
            ((unsigned long long)__float_as_uint(metric) << 32) |
            (unsigned long long)(0xFFFFFFFFu - (unsigned)a);
        // replace current min if better
        int mp = 0;
        unsigned long long mv = keys[0];
        #pragma unroll
        for (int i = 1; i < TOPK; ++i)
            if (keys[i] < mv) { mv = keys[i]; mp = i; }
        if (key > mv) keys[mp] = key;
    }

    // 13 rounds of block-wide argmax over the union of local lists
    for (int r = 0; r < TOPK; ++r) {
        unsigned long long lm = keys[0];
        #pragma unroll
        for (int i = 1; i < TOPK; ++i) lm = umax64(lm, keys[i]);
        sKeys[tid] = lm;
        __syncthreads();
        for (int s = 128; s > 0; s >>= 1) {
            if (tid < s) sKeys[tid] = umax64(sKeys[tid], sKeys[tid + s]);
            __syncthreads();
        }
        unsigned long long win = sKeys[0];
        if (tid == 0)
            sSel[r] = (int)(0xFFFFFFFFu - (unsigned)(win & 0xFFFFFFFFull));
        #pragma unroll
        for (int i = 0; i < TOPK; ++i)
            if (keys[i] == win) keys[i] = 0ull;
        __syncthreads();
    }

    // mask_pos = mask_topk * mask_in_gts * mask_gt (indices distinct here)
    if (tid < TOPK) {
        int a = sSel[tid];
        float x = anc[2 * a], y = anc[2 * a + 1];
        float d = fminf(fminf(x - gx1, y - gy1), fminf(gx2 - x, gy2 - y));
        if (d > 1e-9f) maskPos[(size_t)bg * na + a] = 1;
    }
}

// ---------------------------------------------------------------------------
// Kernel C: per-anchor multi-assignment resolution + pos_align/pos_over.
// GT tables staged into LDS with CDNA5 async global->LDS B128 loads.
// ---------------------------------------------------------------------------
__global__ void resolve_kernel(const float* __restrict__ pdScores,
                               const float* __restrict__ pdBoxes,
                               const float* __restrict__ anc,
                               const int* __restrict__ gtLabels,
                               const float* __restrict__ gtBoxes,
                               const float* __restrict__ maskGt,
                               const unsigned char* __restrict__ maskPos,
                               int* __restrict__ tgArr,
                               int* __restrict__ fgArr,
                               float* __restrict__ metricSel,
                               float* __restrict__ overSel,
                               unsigned int* __restrict__ posAlign,
                               unsigned int* __restrict__ posOver,
                               int na) {
    __shared__ float sGB[NMB * 4];
    __shared__ int   sLB[NMB];
    __shared__ float sMG[NMB];

    const int b = blockIdx.y;
    const int tid = threadIdx.x;

#if USE_ASYNC_LDS && defined(__AMDGCN__)
    if (tid < 32) {
        // gt_bboxes: 1024B = 2 x (32 lanes x 16B)
        const char* srcB = (const char*)(gtBoxes + (size_t)b * NMB * 4) + tid * 16;
        unsigned dstB = (unsigned)(size_t)&sGB[tid * 4];
        asm volatile("global_load_async_to_lds_b128 %0, %1, off"
                     :: "v"(dstB), "v"(srcB) : "memory");
        const char* srcB2 = srcB + 512;
        unsigned dstB2 = dstB + 512;
        asm volatile("global_load_async_to_lds_b128 %0, %1, off"
                     :: "v"(dstB2), "v"(srcB2) : "memory");
        if (tid < 16) {
            const char* srcL = (const char*)(gtLabels + (size_t)b * NMB) + tid * 16;
            unsigned dstL = (unsigned)(size_t)&sLB[tid * 4];
            asm volatile("global_load_async_to_lds_b128 %0, %1, off"
                         :: "v"(dstL), "v"(srcL) : "memory");
            const char* srcM = (const char*)(maskGt + (size_t)b * NMB) + tid * 16;
            unsigned dstM = (unsigned)(size_t)&sMG[tid * 4];
            asm volatile("global_load_async_to_lds_b128 %0, %1, off"
                         :: "v"(dstM), "v"(srcM) : "memory");
        }
        asm volatile("s_wait_asynccnt 0" ::: "memory");
    }
#else
    for (int i = tid; i < NMB * 4; i += blockDim.x) sGB[i] = gtBoxes[(size_t)b * NMB * 4 + i];
    for (int i = tid; i < NMB; i += blockDim.x) {
        sLB[i] = gtLabels[b * NMB + i];
        sMG[i] = maskGt[b * NMB + i];
    }
#endif
    __syncthreads();

    const int a = blockIdx.x * 256 + tid;
    if (a >= na) return;

    const float x = anc[2 * a], y = anc[2 * a + 1];
    const float* pb = pdBoxes + ((size_t)b * na + a) * 4;
    const float px1 = pb[0], py1 = pb[1], px2 = pb[2], py2 = pb[3];

    const unsigned char* mrow = maskPos + (size_t)b * NMB * na + a;
    int fg = 0, tg = 0, found = 0;
    for (int g = 0; g < NMB; ++g) {
        int m = mrow[(size_t)g * na];
        fg += m;
        if (m && !found) { tg = g; found = 1; }
    }

    if (fg > 1) {
        // argmax over g of masked overlaps (ties -> lowest g)
        float best = -1.0f;
        int bi = 0;
        for (int g = 0; g < NMB; ++g) {
            float gx1 = sGB[4 * g], gy1 = sGB[4 * g + 1];
            float gx2 = sGB[4 * g + 2], gy2 = sGB[4 * g + 3];
            float d = fminf(fminf(x - gx1, y - gy1), fminf(gx2 - x, gy2 - y));
            float ov = 0.f;
            if (d > 1e-9f && sMG[g] > 0.f) {
                float ci = ciou_box(gx1, gy1, gx2, gy2, px1, py1, px2, py2);
                ov = fmaxf(ci, 0.f);
            }
            if (ov > best) { best = ov; bi = g; }
        }
        tg = bi;
    }
    const int fgOut = (fg > 0) ? 1 : 0;

    float ms = 0.f, os = 0.f;
    if (fgOut) {
        float gx1 = sGB[4 * tg], gy1 = sGB[4 * tg + 1];
        float gx2 = sGB[4 * tg + 2], gy2 = sGB[4 * tg + 3];
        float d = fminf(fminf(x - gx1, y - gy1), fminf(gx2 - x, gy2 - y));
        if (d > 1e-9f && sMG[tg] > 0.f) {
            float ci = ciou_box(gx1, gy1, gx2, gy2, px1, py1, px2, py2);
            float ov = fmaxf(ci, 0.f);
            float sc = pdScores[((size_t)b * na + a) * NCLS + sLB[tg]];
            float ov2 = ov * ov;
            os = ov;
            ms = sc * (ov2 * ov2 * ov2);
        }
        atomicMax(&posAlign[b * NMB + tg], __float_as_uint(ms));
        atomicMax(&posOver[b * NMB + tg], __float_as_uint(os));
    }

    const size_t idx = (size_t)b * na + a;
    tgArr[idx] = tg;
    fgArr[idx] = fgOut;
    metricSel[idx] = ms;
    overSel[idx] = os;
}

// ---------------------------------------------------------------------------
// Kernel D: norm + all five outputs (concatenated flat, float-cast).
// ---------------------------------------------------------------------------
__global__ void finalize_kernel(const int* __restrict__ gtLabels,
                                const float* __restrict__ gtBoxes,
                                const int* __restrict__ tgArr,
                                const int* __restrict__ fgArr,
                                const float* __restrict__ metricSel,
                                const unsigned int* __restrict__ posAlign,
                                const unsigned int* __restrict__ posOver,
                                float* __restrict__ out,
                                int na, int bs) {
    const int b = blockIdx.y;
    const int a = blockIdx.x * 256 + threadIdx.x;
    if (a >= na) return;
    const size_t L = (size_t)bs * na;
    const size_t idx = (size_t)b * na + a;

    const int tg = tgArr[idx];
    const int fg = fgArr[idx];
    const float pa = __uint_as_float(posAlign[b * NMB + tg]);
    const float po = __uint_as_float(posOver[b * NMB + tg]);
    const float norm = fg ? (metricSel[idx] * po / (pa + 1e-9f)) : 0.f;

    int lbl = gtLabels[b * NMB + tg];
    if (lbl < 0) lbl = 0;

    // target_labels
    out[idx] = (float)lbl;
    // target_bboxes
    const float* gb = gtBoxes + ((size_t)b * NMB + tg) * 4;
    float* ob = out + L + idx * 4;
    ob[0] = gb[0]; ob[1] = gb[1]; ob[2] = gb[2]; ob[3] = gb[3];
    // target_scores (one-hot * norm, zero when !fg since norm==0)
    float* srow = out + 5 * L + idx * NCLS;
    for (int c = 0; c < NCLS; ++c) srow[c] = 0.f;
    if (fg) srow[lbl] = norm;
    // fg_mask
    out[85 * L + idx] = fg ? 1.f : 0.f;
    // target_gt_idx
    out[86 * L + idx] = (float)tg;
}

// ---------------------------------------------------------------------------
extern "C" void kernel_launch(void* const* d_in, const int* in_sizes, int n_in,
                              void* d_out, int out_size, void* d_ws, size_t ws_size,
                              hipStream_t stream) {
    const float* pdScores = (const float*)d_in[0];
    const float* pdBoxes  = (const float*)d_in[1];
    const float* anc      = (const float*)d_in[2];
    const int*   gtLabels = (const int*)d_in[3];
    const float* gtBoxes  = (const float*)d_in[4];
    const float* maskGt   = (const float*)d_in[5];

    const int na = in_sizes[2] / 2;
    const int bs = in_sizes[5] / NMB;
    const size_t L = (size_t)bs * na;

    auto alignUp = [](size_t x) { return (x + 255) & ~(size_t)255; };
    char* base = (char*)d_ws;
    size_t off = 0;
    unsigned char* maskPos = (unsigned char*)(base + off);
    const size_t maskBytes = (size_t)bs * NMB * na;
    off += alignUp(maskBytes);
    int* tgArr = (int*)(base + off);         off += alignUp(L * sizeof(int));
    int* fgArr = (int*)(base + off);         off += alignUp(L * sizeof(int));
    float* metricSel = (float*)(base + off); off += alignUp(L * sizeof(float));
    float* overSel   = (float*)(base + off); off += alignUp(L * sizeof(float));
    unsigned int* posAlign = (unsigned int*)(base + off);
    off += alignUp((size_t)bs * NMB * sizeof(unsigned int));
    unsigned int* posOver = (unsigned int*)(base + off);
    off += alignUp((size_t)bs * NMB * sizeof(unsigned int));

    float* scoresMat = nullptr;
    const size_t scoresBytes = (size_t)bs * NMB * na * sizeof(float);
    if ((na % 16) == 0 && ws_size >= off + scoresBytes) {
        scoresMat = (float*)(base + off);
    }

    hipMemsetAsync(maskPos, 0, maskBytes, stream);
    hipMemsetAsync(posAlign, 0, (size_t)bs * NMB * sizeof(unsigned int), stream);
    hipMemsetAsync(posOver, 0, (size_t)bs * NMB * sizeof(unsigned int), stream);

    if (scoresMat) {
        const int aTiles = na / 16;
        const int waves = bs * (NMB / 16) * aTiles;
        const int blocks = (waves + 3) / 4;  // 4 waves (128 threads) per block
        scores_wmma_kernel<<<blocks, 128, 0, stream>>>(pdScores, gtLabels,
                                                       scoresMat, na, bs);
    }

    topk_kernel<<<bs * NMB, 256, 0, stream>>>(pdScores, pdBoxes, anc, gtLabels,
                                              gtBoxes, maskGt, scoresMat,
                                              maskPos, na);

    dim3 grid((na + 255) / 256, bs);
    resolve_kernel<<<grid, 256, 0, stream>>>(pdScores, pdBoxes, anc, gtLabels,
                                             gtBoxes, maskGt, maskPos,
                                             tgArr, fgArr, metricSel, overSel,
                                             posAlign, posOver, na);

    finalize_kernel<<<grid, 256, 0, stream>>>(gtLabels, gtBoxes, tgArr, fgArr,
                                              metricSel, posAlign, posOver,
                                              (float*)d_out, na, bs);
}